// DCTFeatureExtractor_17600775979597
// MI455X (gfx1250) — compile-verified
//
#include <hip/hip_runtime.h>

typedef float v2f __attribute__((ext_vector_type(2)));
typedef float v8f __attribute__((ext_vector_type(8)));

#define NIMG 32
#define KDIM 16384   // 64*64 blocks * 4 features
#define NDIM 512

// ---------------------------------------------------------------------------
// Stage 1: per-8x8-block feature extraction.
// feat_m = (1/8) * sum_l colsum[l] * cos(pi*m*l/4),  m = 0..3
// One thread per block; 131072 threads total.
// ---------------------------------------------------------------------------
__global__ __launch_bounds__(256) void dct_feat_kernel(const float* __restrict__ x,
                                                       float* __restrict__ feats)
{
    int t = blockIdx.x * 256 + threadIdx.x;   // 0 .. 131071
    int j = t & 63;                            // block col
    int i = (t >> 6) & 63;                     // block row
    int b = t >> 12;                           // image

    // channel 0 of image b, block (i, j)
    const float* p = x + (size_t)b * 3u * 512u * 512u + (size_t)(i * 8) * 512u + (size_t)(j * 8);

    float c0 = 0.f, c1 = 0.f, c2 = 0.f, c3 = 0.f;
    float c4 = 0.f, c5 = 0.f, c6 = 0.f, c7 = 0.f;
#pragma unroll
    for (int r = 0; r < 8; ++r) {
        const float4* q = (const float4*)(p + (size_t)r * 512u);
        float4 v0 = q[0];
        float4 v1 = q[1];
        c0 += v0.x; c1 += v0.y; c2 += v0.z; c3 += v0.w;
        c4 += v1.x; c5 += v1.y; c6 += v1.z; c7 += v1.w;
    }

    const float s = 0.70710678118654752f;  // sqrt(2)/2
    float f0 = 0.125f * (c0 + c1 + c2 + c3 + c4 + c5 + c6 + c7);
    float f1 = 0.125f * (c0 + s * c1 - s * c3 - c4 - s * c5 + s * c7);
    float f2 = 0.125f * (c0 - c2 + c4 - c6);
    float f3 = 0.125f * (c0 - s * c1 + s * c3 - c4 + s * c5 - s * c7);

    // feats[b][(i*64 + j)*4 + m]  ==  flat index 4*t + m
    float4 o; o.x = f0; o.y = f1; o.z = f2; o.w = f3;
    ((float4*)feats)[t] = o;
}

// ---------------------------------------------------------------------------
// Stage 2: out = feats(32x16384) @ weight(512x16384)^T  via V_WMMA_F32_16X16X4_F32.
// One wave per (mt, nt, kt): 2 x 32 x 16 = 1024 waves; each wave owns a 16x16
// C tile over a 1024-wide K slice and writes a partial to workspace.
//
// A frag (16x4 f32, ISA layout): lane L holds row M = L%16,
//   vgpr g holds K = g + 2*(L/16)  -> per lane a float2 at (k4*4 + 2*(L/16)).
// B frag mirrors A with N in place of M (B column n = weight row n, contiguous K).
// ---------------------------------------------------------------------------
__global__ __launch_bounds__(256) void wmma_gemm_kernel(const float* __restrict__ feats,
                                                        const float* __restrict__ weight,
                                                        float* __restrict__ partials)
{
    int lane = threadIdx.x & 31;
    int wave = (blockIdx.x * 256 + threadIdx.x) >> 5;  // 0 .. 1023
    int kt = wave & 15;
    int nt = (wave >> 4) & 31;
    int mt = wave >> 9;

    int lm = lane & 15;        // row-within-tile (A) / col-within-tile (B)
    int h  = lane >> 4;        // lane half selects K pairs {0,1} vs {2,3}

    const float* ap = feats  + (size_t)(mt * 16 + lm) * KDIM + 2 * h;
    const float* bp = weight + (size_t)(nt * 16 + lm) * KDIM + 2 * h;

    v8f c = {};                // zero accumulator
    const int kbase = kt * 1024;

    for (int ko = 0; ko < 16; ++ko) {
        int off0 = kbase + ko * 64;
        // stream the weight row ~2KB ahead (global_prefetch_b8)
        __builtin_prefetch(bp + off0 + 512, 0, 0);
#pragma unroll
        for (int ki = 0; ki < 16; ++ki) {
            int off = off0 + ki * 4;
            v2f a = *(const v2f*)(ap + off);
            v2f b = *(const v2f*)(bp + off);
            c = __builtin_amdgcn_wmma_f32_16x16x4_f32(
                    /*neg_a=*/false, a, /*neg_b=*/false, b,
                    /*c_mod=*/(short)0, c, /*reuse_a=*/false, /*reuse_b=*/false);
        }
    }

    // partials[tile = mt*32+nt][kt][r*32 + lane]
    float* outp = partials + ((size_t)(mt * 32 + nt) * 16 + kt) * 256 + lane;
#pragma unroll
    for (int r = 0; r < 8; ++r) outp[r * 32] = c[r];
}

// ---------------------------------------------------------------------------
// Stage 3: reduce the 16 K-partials per tile, add bias, scatter to d_out.
// C tile layout: VGPR r, lanes 0-15 -> M=r, N=lane; lanes 16-31 -> M=r+8, N=lane-16.
// ---------------------------------------------------------------------------
__global__ __launch_bounds__(256) void reduce_kernel(const float* __restrict__ partials,
                                                     const float* __restrict__ bias,
                                                     float* __restrict__ out)
{
    int tile = blockIdx.x;     // 0 .. 63
    int idx  = threadIdx.x;    // 0 .. 255  == r*32 + lane
    int nt = tile & 31;
    int mt = tile >> 5;

    const float* p = partials + (size_t)tile * 16u * 256u + idx;
    float sum = 0.f;
#pragma unroll
    for (int kt = 0; kt < 16; ++kt) sum += p[kt * 256];

    int r    = idx >> 5;
    int lane = idx & 31;
    int M = r + 8 * (lane >> 4);
    int N = lane & 15;
    int row = mt * 16 + M;     // batch index 0..31
    int col = nt * 16 + N;     // output feature 0..511
    out[(size_t)row * NDIM + col] = sum + bias[col];
}

extern "C" void kernel_launch(void* const* d_in, const int* in_sizes, int n_in,
                              void* d_out, int out_size, void* d_ws, size_t ws_size,
                              hipStream_t stream)
{
    const float* imgs   = (const float*)d_in[0];  // (32, 3, 512, 512) f32
    const float* weight = (const float*)d_in[1];  // (512, 16384) f32
    const float* bias   = (const float*)d_in[2];  // (512,) f32
    (void)in_sizes; (void)n_in; (void)out_size; (void)ws_size;

    float* feats    = (float*)d_ws;                                   // 32*16384 f32 = 2 MB
    float* partials = (float*)((char*)d_ws + (size_t)NIMG * KDIM * sizeof(float)); // 64*16*256 f32 = 1 MB

    // Stage 1: 131072 block-threads
    dct_feat_kernel<<<512, 256, 0, stream>>>(imgs, feats);

    // Stage 2: 1024 waves = 128 blocks x 8 waves
    wmma_gemm_kernel<<<128, 256, 0, stream>>>(feats, weight, partials);

    // Stage 3: one block per 16x16 output tile
    reduce_kernel<<<64, 256, 0, stream>>>(partials, bias, (float*)d_out);
}